// MatchTarget_69114613729413
// MI455X (gfx1250) — compile-verified
//
#include <hip/hip_runtime.h>
#include <stdint.h>

#define MT_P 65536
#define MT_B 8
#define MT_G 64
#define MT_IOU_TH 0.5f
#define MT_VAR_C 0.1f
#define MT_VAR_WH 0.2f

// Pointer types matching the async-to-LDS builtin signature:
//   param0: int4* in global (addrspace 1), param1: int4* in LDS (addrspace 3)
typedef int mt_i4 __attribute__((ext_vector_type(4)));
typedef __attribute__((address_space(1))) mt_i4* mt_g4p;
typedef __attribute__((address_space(3))) mt_i4* mt_l4p;

// ---------------------------------------------------------------------------
// IoU between two corner-form boxes given their areas.
// ---------------------------------------------------------------------------
__device__ __forceinline__ float mt_iou(const float4 a, const float4 b,
                                        float area_a, float area_b) {
    float ltx = fmaxf(a.x, b.x), lty = fmaxf(a.y, b.y);
    float rbx = fminf(a.z, b.z), rby = fminf(a.w, b.w);
    float w = fmaxf(rbx - ltx, 0.0f), h = fmaxf(rby - lty, 0.0f);
    float inter = w * h;
    return inter / (area_a + area_b - inter);
}

// ---------------------------------------------------------------------------
// Kernel 1: best prior per (batch, gt). One block per (g, b).
// Packs (iou_bits<<32) | (0xFFFFFFFF - p): max over packed == argmax with
// lowest-prior-index tie-break (== jnp.argmax over axis 0, first occurrence).
// ---------------------------------------------------------------------------
__global__ __launch_bounds__(256)
void mt_best_prior_kernel(const float4* __restrict__ priors,
                          const float4* __restrict__ gt_boxes,
                          unsigned* __restrict__ best_prior) {
    const int g   = blockIdx.x;
    const int b   = blockIdx.y;
    const int tid = threadIdx.x;

    const float4 gb = gt_boxes[b * MT_G + g];           // uniform broadcast
    const float area_b = (gb.z - gb.x) * (gb.w - gb.y);

    unsigned long long best = 0ull;
    for (int p = tid; p < MT_P; p += 256) {
        if (p + 256 < MT_P)
            __builtin_prefetch(&priors[p + 256], 0, 3); // global_prefetch_b8
        float4 pr = priors[p];
        float area_a = (pr.z - pr.x) * (pr.w - pr.y);
        float v = mt_iou(pr, gb, area_a, area_b);
        unsigned long long packed =
            (((unsigned long long)__float_as_uint(v)) << 32) |
            (unsigned long long)(0xFFFFFFFFu - (unsigned)p);
        best = (packed > best) ? packed : best;
    }

    // wave32 shuffle reduction
    #pragma unroll
    for (int off = 16; off > 0; off >>= 1) {
        unsigned long long o = __shfl_down(best, off, 32);
        best = (o > best) ? o : best;
    }

    __shared__ unsigned long long red[8];
    const int wid = tid >> 5, lane = tid & 31;
    if (lane == 0) red[wid] = best;
    __syncthreads();
    if (tid == 0) {
        unsigned long long m = red[0];
        #pragma unroll
        for (int i = 1; i < 8; ++i) m = (red[i] > m) ? red[i] : m;
        best_prior[b * MT_G + g] = 0xFFFFFFFFu - (unsigned)(m & 0xFFFFFFFFull);
    }
}

// ---------------------------------------------------------------------------
// Kernel 2: per-prior match + SSD encode. One thread per (b, p).
// GT boxes staged into LDS via CDNA5 async global->LDS when available.
// ---------------------------------------------------------------------------
__global__ __launch_bounds__(256)
void mt_match_encode_kernel(const float4* __restrict__ priors,
                            const float4* __restrict__ gt_boxes,
                            const int* __restrict__ gt_labels,
                            const unsigned* __restrict__ best_prior,
                            float4* __restrict__ loc,
                            int* __restrict__ conf) {
    const int b   = blockIdx.y;
    const int tid = threadIdx.x;
    const int p   = blockIdx.x * 256 + tid;

    __shared__ float4   sbox[MT_G];
    __shared__ int      slab[MT_G];
    __shared__ unsigned sbp[MT_G];

    if (tid < MT_G) {
#if __has_builtin(__builtin_amdgcn_global_load_async_to_lds_b128)
        // CDNA5 async tensor path: global_load_async_to_lds_b128 (ASYNCcnt)
        __builtin_amdgcn_global_load_async_to_lds_b128(
            (mt_g4p)(gt_boxes + b * MT_G + tid),
            (mt_l4p)&sbox[tid],
            0, 0);
#else
        sbox[tid] = gt_boxes[b * MT_G + tid];
#endif
        slab[tid] = gt_labels[b * MT_G + tid];
        sbp[tid]  = best_prior[b * MT_G + tid];
    }
#if __has_builtin(__builtin_amdgcn_global_load_async_to_lds_b128)
#  if __has_builtin(__builtin_amdgcn_s_wait_asynccnt)
    __builtin_amdgcn_s_wait_asynccnt(0);
#  else
    asm volatile("s_wait_asynccnt 0" ::: "memory");
#  endif
#endif
    __syncthreads();

    const float4 pr = priors[p];
    const float area_a = (pr.z - pr.x) * (pr.w - pr.y);

    // argmax over g, strict '>' keeps first occurrence (jnp.argmax axis=1)
    float best_iou = -1.0f;
    int   best_g   = 0;
    #pragma unroll 8
    for (int g = 0; g < MT_G; ++g) {
        float4 gb = sbox[g];
        float area_b = (gb.z - gb.x) * (gb.w - gb.y);
        float v = mt_iou(pr, gb, area_a, area_b);
        if (v > best_iou) { best_iou = v; best_g = g; }
    }

    // force each GT's best prior to keep that GT (ascending g: last write wins)
    #pragma unroll 8
    for (int g = 0; g < MT_G; ++g) {
        if (sbp[g] == (unsigned)p) { best_g = g; best_iou = 2.0f; }
    }

    const int c = (best_iou < MT_IOU_TH) ? 0 : (slab[best_g] + 1);

    // SSD encode of matched box against center-form prior
    const float4 m  = sbox[best_g];
    const float pcx = (pr.x + pr.z) * 0.5f, pcy = (pr.y + pr.w) * 0.5f;
    const float pw  = pr.z - pr.x,          ph  = pr.w - pr.y;
    const float gcx = (m.x + m.z) * 0.5f,   gcy = (m.y + m.w) * 0.5f;
    const float gw  = m.z - m.x,            gh  = m.w - m.y;

    float4 o;
    o.x = (gcx - pcx) / (MT_VAR_C * pw);
    o.y = (gcy - pcy) / (MT_VAR_C * ph);
    o.z = __logf(gw / pw) * (1.0f / MT_VAR_WH);
    o.w = __logf(gh / ph) * (1.0f / MT_VAR_WH);

    const size_t idx = (size_t)b * MT_P + (size_t)p;
    loc[idx]  = o;
    conf[idx] = c;
}

// ---------------------------------------------------------------------------
// Launch: kernel1 -> ws (512 ints), kernel2 -> loc + conf in d_out.
// d_out layout: [B,P,4] f32 loc, then [B,P] i32 conf (flat, return order).
// ---------------------------------------------------------------------------
extern "C" void kernel_launch(void* const* d_in, const int* in_sizes, int n_in,
                              void* d_out, int out_size, void* d_ws, size_t ws_size,
                              hipStream_t stream) {
    const float4* priors    = (const float4*)d_in[0];   // [P,4] f32
    const float4* gt_boxes  = (const float4*)d_in[1];   // [B,G,4] f32
    const int*    gt_labels = (const int*)d_in[2];      // [B,G] i32

    unsigned* best_prior = (unsigned*)d_ws;             // [B,G] = 2 KB scratch

    float4* loc  = (float4*)d_out;
    int*    conf = (int*)((float*)d_out + (size_t)MT_B * MT_P * 4);

    mt_best_prior_kernel<<<dim3(MT_G, MT_B), 256, 0, stream>>>(
        priors, gt_boxes, best_prior);

    mt_match_encode_kernel<<<dim3(MT_P / 256, MT_B), 256, 0, stream>>>(
        priors, gt_boxes, gt_labels, best_prior, loc, conf);
}